// MyGATConv_18691697672390
// MI455X (gfx1250) — compile-verified
//
#include <hip/hip_runtime.h>
#include <math.h>

#define N_NODES 8192
#define DIN 512
#define DOUT 256
#define MAXE 4096

typedef __attribute__((ext_vector_type(2))) float v2f;
typedef __attribute__((ext_vector_type(4))) float v4f;
typedef __attribute__((ext_vector_type(8))) float v8f;

// ---------------------------------------------------------------------------
// Kernel 1: h = elu(x) @ W1 + b1   (fp32 WMMA 16x16x4)
// block = 256 threads (8 waves). Tile: 128 rows x 64 cols, K-slab = 32.
// Each wave owns a 16-row strip and 4 column sub-tiles (16 cols each).
// ---------------------------------------------------------------------------
__global__ __launch_bounds__(256)
void gemm_h_kernel(const float* __restrict__ x, const float* __restrict__ W1,
                   const float* __restrict__ b1, float* __restrict__ h) {
    __shared__ float sX[128][34];   // [m][k], pad 2 to spread banks
    __shared__ float sW[64][34];    // [n][k] (W1 slab transposed), pad 2

    const int t     = threadIdx.x;
    const int w     = t >> 5;
    const int lane  = t & 31;
    const int R     = blockIdx.x * 128;
    const int Cb    = blockIdx.y * 64;
    const int mloc  = w * 16 + (lane & 15);
    const int klane = (lane >> 4) * 2;      // lanes 0-15: K=0,1 ; 16-31: K=2,3
    const int nlane = lane & 15;

    v8f acc[4] = {};

    for (int k0 = 0; k0 < DIN; k0 += 32) {
        // cooperative load of elu(x) slab: 128 x 32
        #pragma unroll
        for (int r = 0; r < 16; ++r) {
            int idx = r * 256 + t;
            int m = idx >> 5, k = idx & 31;
            float v = x[(size_t)(R + m) * DIN + (k0 + k)];
            sX[m][k] = v > 0.0f ? v : expm1f(v);   // jax.nn.elu, alpha=1
        }
        // cooperative load of W1 slab (transposed to [n][k]): 32 x 64
        #pragma unroll
        for (int r = 0; r < 8; ++r) {
            int idx = r * 256 + t;
            int k = idx >> 6, n = idx & 63;
            sW[n][k] = W1[(size_t)(k0 + k) * DOUT + (Cb + n)];
        }
        __syncthreads();

        #pragma unroll
        for (int kk = 0; kk < 32; kk += 4) {
            v2f a = *(const v2f*)&sX[mloc][kk + klane];
            #pragma unroll
            for (int nt = 0; nt < 4; ++nt) {
                v2f b = *(const v2f*)&sW[nt * 16 + nlane][kk + klane];
                acc[nt] = __builtin_amdgcn_wmma_f32_16x16x4_f32(
                    false, a, false, b, (short)0, acc[nt], false, false);
            }
        }
        __syncthreads();
    }

    // D layout: VGPR r -> row r (lanes 0-15) / row 8+r (lanes 16-31)
    const int halfsel = lane >> 4;
    #pragma unroll
    for (int nt = 0; nt < 4; ++nt) {
        int col = Cb + nt * 16 + nlane;
        float bias = b1[col];
        #pragma unroll
        for (int r = 0; r < 8; ++r) {
            int row = R + w * 16 + halfsel * 8 + r;
            h[(size_t)row * DOUT + col] = acc[nt][r] + bias;
        }
    }
}

// ---------------------------------------------------------------------------
// Kernel 2: g1 = h @ a1_w + a1_b ; g2 = h @ a2_w + a2_b  (one wave per row)
// ---------------------------------------------------------------------------
__global__ __launch_bounds__(256)
void g_scores_kernel(const float* __restrict__ h,
                     const float* __restrict__ a1w, const float* __restrict__ a1b,
                     const float* __restrict__ a2w, const float* __restrict__ a2b,
                     float* __restrict__ g1, float* __restrict__ g2) {
    const int w = threadIdx.x >> 5, lane = threadIdx.x & 31;
    const int row = blockIdx.x * 8 + w;
    float s1 = 0.f, s2 = 0.f;
    for (int c = lane; c < DOUT; c += 32) {
        float hv = h[(size_t)row * DOUT + c];
        s1 += hv * a1w[c];
        s2 += hv * a2w[c];
    }
    #pragma unroll
    for (int off = 16; off > 0; off >>= 1) {
        s1 += __shfl_xor(s1, off, 32);
        s2 += __shfl_xor(s2, off, 32);
    }
    if (lane == 0) {
        g1[row] = s1 + a1b[0];
        g2[row] = s2 + a2b[0];
    }
}

// ---------------------------------------------------------------------------
// Kernel 3: fused masked softmax + aggregation. One block per node i.
// Streams adj row with non-temporal 128-bit loads (the HBM-bound part:
// 256 MB total ~= 11.5 us at 23.3 TB/s), caches edge indices in LDS
// (~32/row), softmax over edges, then per-column weighted gather of h
// (8 MB, L2-resident on the 192 MB L2).
// ---------------------------------------------------------------------------
__global__ __launch_bounds__(256)
void attn_agg_kernel(const float* __restrict__ adj, const float* __restrict__ h,
                     const float* __restrict__ g1, const float* __restrict__ g2,
                     float* __restrict__ out) {
    __shared__ int   s_idx[MAXE];
    __shared__ float s_w[MAXE];
    __shared__ int   s_cnt;
    __shared__ float s_red[8];
    __shared__ float s_bc;

    const int i = blockIdx.x;
    const int t = threadIdx.x;
    const int w = t >> 5, lane = t & 31;

    if (t == 0) s_cnt = 0;
    __syncthreads();

    // stream adjacency row as float4 (b128, non-temporal), append edge indices
    const v4f* arow4 = (const v4f*)(adj + (size_t)i * N_NODES);
    #pragma unroll
    for (int q = 0; q < N_NODES / (4 * 256); ++q) {      // 8 iterations
        int j4 = q * 256 + t;
        v4f a = __builtin_nontemporal_load(&arow4[j4]);
        #pragma unroll
        for (int c = 0; c < 4; ++c) {
            if (a[c] != 0.0f) {
                int pos = atomicAdd(&s_cnt, 1);
                if (pos < MAXE) s_idx[pos] = j4 * 4 + c;
            }
        }
    }
    __syncthreads();
    int cnt = s_cnt; if (cnt > MAXE) cnt = MAXE;

    if (cnt == 0) {
        // all-(-inf) row -> uniform softmax -> column mean of h
        float acc = 0.f;
        for (int j = 0; j < N_NODES; ++j) acc += h[(size_t)j * DOUT + t];
        out[(size_t)i * DOUT + t] = acc / (float)N_NODES;
        return;
    }

    const float g2i = g2[i];

    // pass 1: leaky_relu scores + block max
    float lmax = -INFINITY;
    for (int e = t; e < cnt; e += 256) {
        float v = g2i + g1[s_idx[e]];
        v = v >= 0.f ? v : 0.2f * v;          // leaky_relu slope 0.2
        s_w[e] = v;
        lmax = fmaxf(lmax, v);
    }
    #pragma unroll
    for (int off = 16; off > 0; off >>= 1) lmax = fmaxf(lmax, __shfl_xor(lmax, off, 32));
    if (lane == 0) s_red[w] = lmax;
    __syncthreads();
    if (t == 0) {
        float m = s_red[0];
        #pragma unroll
        for (int q = 1; q < 8; ++q) m = fmaxf(m, s_red[q]);
        s_bc = m;
    }
    __syncthreads();
    const float m = s_bc;

    // pass 2: exp + block sum
    float lsum = 0.f;
    for (int e = t; e < cnt; e += 256) {
        float wv = expf(s_w[e] - m);
        s_w[e] = wv;
        lsum += wv;
    }
    #pragma unroll
    for (int off = 16; off > 0; off >>= 1) lsum += __shfl_xor(lsum, off, 32);
    if (lane == 0) s_red[w] = lsum;
    __syncthreads();
    if (t == 0) {
        float z = 0.f;
        #pragma unroll
        for (int q = 0; q < 8; ++q) z += s_red[q];
        s_bc = z;
    }
    __syncthreads();
    const float invZ = 1.0f / s_bc;

    // aggregation: thread t owns output column t (DOUT == blockDim.x)
    float acc = 0.f;
    const int col = t;
    for (int e = 0; e < cnt; ++e) {
        int j = s_idx[e];
        if (e + 1 < cnt)
            __builtin_prefetch(&h[(size_t)s_idx[e + 1] * DOUT + col], 0, 0);
        acc += s_w[e] * h[(size_t)j * DOUT + col];
    }
    out[(size_t)i * DOUT + col] = acc * invZ;
}

// ---------------------------------------------------------------------------
extern "C" void kernel_launch(void* const* d_in, const int* in_sizes, int n_in,
                              void* d_out, int out_size, void* d_ws, size_t ws_size,
                              hipStream_t stream) {
    const float* x   = (const float*)d_in[0];
    const float* adj = (const float*)d_in[1];
    const float* W1  = (const float*)d_in[2];
    const float* b1  = (const float*)d_in[3];
    const float* a1w = (const float*)d_in[4];
    const float* a1b = (const float*)d_in[5];
    const float* a2w = (const float*)d_in[6];
    const float* a2b = (const float*)d_in[7];
    float* out = (float*)d_out;

    float* h  = (float*)d_ws;                       // [N, DOUT]
    float* g1 = h + (size_t)N_NODES * DOUT;         // [N]
    float* g2 = g1 + N_NODES;                       // [N]

    dim3 grid1(N_NODES / 128, DOUT / 64);
    gemm_h_kernel<<<grid1, 256, 0, stream>>>(x, W1, b1, h);
    g_scores_kernel<<<N_NODES / 8, 256, 0, stream>>>(h, a1w, a1b, a2w, a2b, g1, g2);
    attn_agg_kernel<<<N_NODES, 256, 0, stream>>>(adj, h, g1, g2, out);
}